// GAT_48215302865251
// MI455X (gfx1250) — compile-verified
//
#include <hip/hip_runtime.h>

typedef float v2f __attribute__((ext_vector_type(2)));
typedef float v8f __attribute__((ext_vector_type(8)));

#define NPG 100
#define NUM_GRAPHS 500
#define NNODES (NPG * NUM_GRAPHS)
#define DEG 32
#define IN_DIM 100
#define HID 64
#define HC 128
#define STAT_BLOCKS 125

__device__ __forceinline__ float lrelu(float v) { return v > 0.f ? v : 0.2f * v; }

// ---------------- WMMA fp32 strip GEMM: C[M,N] = A[M,K] @ B[K,N] (+bias)(+lrelu) ------------
// One wave per 16-row strip of C, computing all NT = N/16 column tiles at once so the A
// fragment is loaded from VMEM once per k-step and reused NT times (B is a tiny L2-resident
// weight matrix).  V_WMMA_F32_16X16X4_F32 keeps full fp32 precision: these GEMMs are ~4 GFLOP
// total and nowhere near the bottleneck (the edge gather phase is), so dropping to f16 buys
// nothing on MI455X.
// A 16x4 layout (ISA 7.12.2): lanes 0-15 hold K={0,1}, lanes 16-31 hold K={2,3}, M=lane&15.
// B 4x16 mirrors A; C/D: VGPR r holds rows {r, r+8}, col = lane&15.
template <int NT, int EPI>
__global__ void __launch_bounds__(256) wmma_gemm_strip(
    const float* __restrict__ A, const float* __restrict__ B,
    const float* __restrict__ bias, float* __restrict__ C,
    int M, int K) {
  const int N = NT * 16;
  const int lane = threadIdx.x & 31;
  const int wave = blockIdx.x * (blockDim.x >> 5) + (threadIdx.x >> 5);
  if (wave >= (M >> 4)) return;                 // wave-uniform: EXEC stays all-ones
  const int half  = lane >> 4;                  // 0: lanes 0-15, 1: lanes 16-31
  const int idx16 = lane & 15;
  const int kb    = half << 1;                  // K sub-offset 0 or 2
  const float* Arow = A + (size_t)(wave * 16 + idx16) * K;
  const float* Bcol = B + idx16;
  v8f acc[NT];
#pragma unroll
  for (int t = 0; t < NT; ++t) acc[t] = v8f{};
  for (int k = 0; k < K; k += 4) {
    const v2f a = *(const v2f*)(Arow + k + kb); // 8B aligned: row strides & k are even
    const float* Bk = Bcol + (size_t)(k + kb) * N;
#pragma unroll
    for (int t = 0; t < NT; ++t) {
      v2f b;
      b.x = Bk[t * 16];
      b.y = Bk[t * 16 + N];
      acc[t] = __builtin_amdgcn_wmma_f32_16x16x4_f32(false, a, false, b, (short)0, acc[t],
                                                     false, false);
    }
  }
  const int row0 = wave * 16 + (half << 3);
#pragma unroll
  for (int t = 0; t < NT; ++t) {
    const int col = t * 16 + idx16;
    float bv = 0.f;
    if constexpr (EPI >= 1) bv = bias[col];
#pragma unroll
    for (int r = 0; r < 8; ++r) {
      float v = acc[t][r] + bv;
      if constexpr (EPI >= 2) v = lrelu(v);
      C[(size_t)(row0 + r) * N + col] = v;
    }
  }
}

// ---------------- per-node attention logits: ssrc/sdst[n][h] = <xp[n,h,:], a_{s,d}[h,:]> ---
__global__ void att_scores(const float* __restrict__ xp, const float* __restrict__ a_s,
                           const float* __restrict__ a_d, float* __restrict__ ssrc,
                           float* __restrict__ sdst, int n) {
  int t = blockIdx.x * blockDim.x + threadIdx.x;   // t = node*2 + head
  if (t >= n * 2) return;
  int node = t >> 1, h = t & 1;
  const float* row = xp + (size_t)node * HC + h * HID;
  const float* vs = a_s + h * HID;
  const float* vd = a_d + h * HID;
  float s1 = 0.f, s2 = 0.f;
#pragma unroll 8
  for (int c = 0; c < HID; ++c) { s1 += row[c] * vs[c]; s2 += row[c] * vd[c]; }
  ssrc[t] = s1;
  sdst[t] = s2;
}

// ---------------- per-dst edge softmax + weighted gather-sum --------------------------------
// dst layout is repeat(arange(N), 32) + appended self-loops, so node i owns edges
// [i*32, i*32+32) plus self-loop src=i.  One block of 128 threads per node; the 33-row
// gather of xp[src] is the true bottleneck of the whole net (L2-resident, ~33x row reuse).
__global__ void __launch_bounds__(128) gat_aggregate(
    const float* __restrict__ xp, const float* __restrict__ ssrc,
    const float* __restrict__ sdst, const int* __restrict__ esrc,
    const float* __restrict__ bias, float* __restrict__ out) {
  const int i = blockIdx.x;
  const int tid = threadIdx.x;                 // 128 = HC channels
  __shared__ int   s_src[DEG + 1];
  __shared__ float s_al[2][DEG + 1];
  if (tid <= DEG) {
    int s = (tid < DEG) ? esrc[(size_t)i * DEG + tid] : i;
    s_src[tid] = s;
    s_al[0][tid] = lrelu(ssrc[2 * s]     + sdst[2 * i]);
    s_al[1][tid] = lrelu(ssrc[2 * s + 1] + sdst[2 * i + 1]);
  }
  __syncthreads();
  if (tid < 2) {                               // per-head softmax over the 33 incoming edges
    float m = -3.4e38f;
    for (int j = 0; j <= DEG; ++j) m = fmaxf(m, s_al[tid][j]);
    float sum = 0.f;
    for (int j = 0; j <= DEG; ++j) {
      float e = __expf(s_al[tid][j] - m);
      s_al[tid][j] = e;
      sum += e;
    }
    float inv = 1.f / sum;
    for (int j = 0; j <= DEG; ++j) s_al[tid][j] *= inv;
  }
  __syncthreads();
  const int h = tid >> 6;
  float acc = 0.f;
#pragma unroll 4
  for (int j = 0; j <= DEG; ++j)               // coalesced 128-float row gathers (L2 hot)
    acc += s_al[h][j] * xp[(size_t)s_src[j] * HC + tid];
  out[(size_t)i * HC + tid] = acc + bias[tid];
}

// ---------------- deterministic BN column stats (two stage, no float atomics) ---------------
__global__ void __launch_bounds__(256) col_stats(const float* __restrict__ z,
                                                 float* __restrict__ psum,
                                                 float* __restrict__ psq, int n, int rpb) {
  const int c = threadIdx.x & 63;
  const int r0 = threadIdx.x >> 6;             // 0..3 (4 rows in flight)
  const int base = blockIdx.x * rpb;
  const int end = (base + rpb < n) ? base + rpb : n;
  float s = 0.f, q = 0.f;
  for (int r = base + r0; r < end; r += 4) {
    float v = z[(size_t)r * HID + c];
    s += v;
    q += v * v;
  }
  __shared__ float sh[256], shq[256];
  sh[threadIdx.x] = s;
  shq[threadIdx.x] = q;
  __syncthreads();
  if (r0 == 0) {
    psum[blockIdx.x * HID + c] = sh[c] + sh[64 + c] + sh[128 + c] + sh[192 + c];
    psq[blockIdx.x * HID + c]  = shq[c] + shq[64 + c] + shq[128 + c] + shq[192 + c];
  }
}

__global__ void bn_finalize(const float* __restrict__ psum, const float* __restrict__ psq,
                            const float* __restrict__ g, const float* __restrict__ b,
                            float* __restrict__ scale, float* __restrict__ shift,
                            int n, int nblk) {
  int c = threadIdx.x;
  if (c >= HID) return;
  float s = 0.f, q = 0.f;
  for (int k = 0; k < nblk; ++k) { s += psum[k * HID + c]; q += psq[k * HID + c]; }
  float mean = s / n;
  float var = q / n - mean * mean;              // biased var, matches z.var(0)
  float sc = g[c] * rsqrtf(var + 1e-5f);
  scale[c] = sc;
  shift[c] = b[c] - mean * sc;
}

__global__ void bn_apply(float* __restrict__ z, const float* __restrict__ scale,
                         const float* __restrict__ shift, int total) {
  int idx = blockIdx.x * blockDim.x + threadIdx.x;
  if (idx >= total) return;
  int c = idx & (HID - 1);
  z[idx] = z[idx] * scale[c] + shift[c];
}

// ---------------- global mean pool (batch = arange(N)//100, contiguous groups) --------------
__global__ void pool_mean(const float* __restrict__ z, float* __restrict__ pooled) {
  int g = blockIdx.x, c = threadIdx.x;
  float s = 0.f;
  for (int r = 0; r < NPG; ++r) s += z[(size_t)(g * NPG + r) * HID + c];
  pooled[g * HID + c] = s * (1.0f / NPG);
}

// ---------------- MLP head: one block per graph, staged through LDS -------------------------
__global__ void __launch_bounds__(64) mlp_head(
    const float* __restrict__ pooled,
    const float* __restrict__ f1W, const float* __restrict__ f1b,
    const float* __restrict__ f2W, const float* __restrict__ f2b,
    const float* __restrict__ f3W, const float* __restrict__ f3b,
    float* __restrict__ out) {
  int g = blockIdx.x, c = threadIdx.x;
  __shared__ float in[64], h1[64], h2[32];
  in[c] = pooled[g * 64 + c];
  __syncthreads();
  float a = f1b[c];
  for (int k = 0; k < 64; ++k) a += in[k] * f1W[k * 64 + c];
  h1[c] = lrelu(a);
  __syncthreads();
  if (c < 32) {
    float a2 = f2b[c];
    for (int k = 0; k < 64; ++k) a2 += h1[k] * f2W[k * 32 + c];
    h2[c] = lrelu(a2);
  }
  __syncthreads();
  if (c < 2) {
    float a3 = f3b[c];
    for (int k = 0; k < 32; ++k) a3 += h2[k] * f3W[k * 2 + c];
    out[g * 2 + c] = a3;
  }
}

extern "C" void kernel_launch(void* const* d_in, const int* in_sizes, int n_in,
                              void* d_out, int out_size, void* d_ws, size_t ws_size,
                              hipStream_t stream) {
  const float* x   = (const float*)d_in[0];
  // d_in[1] = edge_attr: computed-but-unused in the reference (alpha-only message)
  const float* W1  = (const float*)d_in[2];
  const float* as1 = (const float*)d_in[3];
  const float* ad1 = (const float*)d_in[4];
  const float* bg1 = (const float*)d_in[5];
  const float* l1W = (const float*)d_in[6];
  const float* l1b = (const float*)d_in[7];
  const float* g1  = (const float*)d_in[8];
  const float* b1  = (const float*)d_in[9];
  const float* W2  = (const float*)d_in[10];
  const float* as2 = (const float*)d_in[11];
  const float* ad2 = (const float*)d_in[12];
  const float* bg2 = (const float*)d_in[13];
  const float* l2W = (const float*)d_in[14];
  const float* l2b = (const float*)d_in[15];
  const float* g2  = (const float*)d_in[16];
  const float* b2  = (const float*)d_in[17];
  const float* f1W = (const float*)d_in[18];
  const float* f1b = (const float*)d_in[19];
  const float* f2W = (const float*)d_in[20];
  const float* f2b = (const float*)d_in[21];
  const float* f3W = (const float*)d_in[22];
  const float* f3b = (const float*)d_in[23];
  const int*   esrc = (const int*)d_in[24];       // edge_index[0] (src); [1] implied by layout
  float* outp = (float*)d_out;

  const int N = NNODES;
  float* ws = (float*)d_ws;
  size_t o = 0;
  float* xp     = ws + o; o += (size_t)N * HC;          // 6.4 M floats
  float* agg    = ws + o; o += (size_t)N * HC;          // 6.4 M
  float* z      = ws + o; o += (size_t)N * HID;         // 3.2 M
  float* ssrc   = ws + o; o += (size_t)N * 2;
  float* sdst   = ws + o; o += (size_t)N * 2;
  float* psum   = ws + o; o += STAT_BLOCKS * HID;
  float* psq    = ws + o; o += STAT_BLOCKS * HID;
  float* bscale = ws + o; o += HID;
  float* bshift = ws + o; o += HID;
  float* pooled = ws + o; o += NUM_GRAPHS * HID;

  const int strips = N / 16;                   // 3125 row-strips, 8 waves per block
  const int gemm_grid = (strips + 7) / 8;

  // ---- layer 1 ----
  wmma_gemm_strip<HC / 16, 0><<<gemm_grid, 256, 0, stream>>>(x, W1, nullptr, xp, N, IN_DIM);
  att_scores<<<(N * 2 + 255) / 256, 256, 0, stream>>>(xp, as1, ad1, ssrc, sdst, N);
  gat_aggregate<<<N, HC, 0, stream>>>(xp, ssrc, sdst, esrc, bg1, agg);
  wmma_gemm_strip<HID / 16, 2><<<gemm_grid, 256, 0, stream>>>(agg, l1W, l1b, z, N, HC);
  col_stats<<<STAT_BLOCKS, 256, 0, stream>>>(z, psum, psq, N, N / STAT_BLOCKS);
  bn_finalize<<<1, 64, 0, stream>>>(psum, psq, g1, b1, bscale, bshift, N, STAT_BLOCKS);
  bn_apply<<<(N * HID + 255) / 256, 256, 0, stream>>>(z, bscale, bshift, N * HID);

  // ---- layer 2 ----
  wmma_gemm_strip<HC / 16, 0><<<gemm_grid, 256, 0, stream>>>(z, W2, nullptr, xp, N, HID);
  att_scores<<<(N * 2 + 255) / 256, 256, 0, stream>>>(xp, as2, ad2, ssrc, sdst, N);
  gat_aggregate<<<N, HC, 0, stream>>>(xp, ssrc, sdst, esrc, bg2, agg);
  wmma_gemm_strip<HID / 16, 2><<<gemm_grid, 256, 0, stream>>>(agg, l2W, l2b, z, N, HC);
  col_stats<<<STAT_BLOCKS, 256, 0, stream>>>(z, psum, psq, N, N / STAT_BLOCKS);
  bn_finalize<<<1, 64, 0, stream>>>(psum, psq, g2, b2, bscale, bshift, N, STAT_BLOCKS);
  bn_apply<<<(N * HID + 255) / 256, 256, 0, stream>>>(z, bscale, bshift, N * HID);

  // ---- pool + head ----
  pool_mean<<<NUM_GRAPHS, HID, 0, stream>>>(z, pooled);
  mlp_head<<<NUM_GRAPHS, 64, 0, stream>>>(pooled, f1W, f1b, f2W, f2b, f3W, f3b, outp);
}